// EdgeConv_29076928594030
// MI455X (gfx1250) — compile-verified
//
#include <hip/hip_runtime.h>
#include <hip/hip_bf16.h>

typedef __attribute__((ext_vector_type(2))) float v2f;
typedef __attribute__((ext_vector_type(8))) float v8f;

#define N_NODES 50000
#define N_EDGES 800000
#define C_IN    128
#define C_OUT   256
#define BN_EPS  1e-5f

// ---------------------------------------------------------------------------
// Phase 0: zero agg + stats buffers (float4 stores)
// ---------------------------------------------------------------------------
__global__ __launch_bounds__(256)
void zero_kernel(float4* __restrict__ p, long long n4) {
    long long i = (long long)blockIdx.x * blockDim.x + threadIdx.x;
    if (i < n4) p[i] = make_float4(0.f, 0.f, 0.f, 0.f);
}

// ---------------------------------------------------------------------------
// Phase 1: A = x @ theta^T, B = x @ phi^T  via V_WMMA_F32_16X16X4_F32.
// One wave per 16(M)x16(N) output tile; K=128 as 32 steps of K=4.
// Layout (ISA 7.12.2):
//   A operand: lane l: {X[l&15][k0+2h], X[l&15][k0+2h+1]}, h = l>>4
//   B operand: lane l: {W[n0+(l&15)][k0+2h], W[n0+(l&15)][k0+2h+1]}
//   C/D:       vgpr r: D[r + 8h][l&15]
// ---------------------------------------------------------------------------
__global__ __launch_bounds__(128)
void node_proj_wmma(const float* __restrict__ x,
                    const float* __restrict__ theta_w,
                    const float* __restrict__ phi_w,
                    float* __restrict__ A,
                    float* __restrict__ B) {
    const int wave = (blockIdx.x * blockDim.x + threadIdx.x) >> 5;
    const int lane = threadIdx.x & 31;
    // total tiles = (N_NODES/16) * (C_OUT/16) = 3125 * 16 = 50000, grid exact
    const int tile_n = wave & 15;
    const int tile_m = wave >> 4;
    const int m0 = tile_m * 16;
    const int n0 = tile_n * 16;
    const int idx  = lane & 15;
    const int half = lane >> 4;

    const float* xrow = x       + (size_t)(m0 + idx) * C_IN + 2 * half;
    const float* trow = theta_w + (size_t)(n0 + idx) * C_IN + 2 * half;
    const float* prow = phi_w   + (size_t)(n0 + idx) * C_IN + 2 * half;

    v8f accT = {};
    v8f accP = {};
#pragma unroll
    for (int s = 0; s < C_IN / 4; ++s) {
        v2f a  = *(const v2f*)(xrow + 4 * s);
        v2f bt = *(const v2f*)(trow + 4 * s);
        v2f bp = *(const v2f*)(prow + 4 * s);
        accT = __builtin_amdgcn_wmma_f32_16x16x4_f32(
            false, a, false, bt, (short)0, accT, false, false);
        accP = __builtin_amdgcn_wmma_f32_16x16x4_f32(
            false, a, false, bp, (short)0, accP, false, false);
    }

#pragma unroll
    for (int r = 0; r < 8; ++r) {
        const int row = m0 + r + 8 * half;
        A[(size_t)row * C_OUT + n0 + idx] = accT[r];
        B[(size_t)row * C_OUT + n0 + idx] = accP[r];
    }
}

// ---------------------------------------------------------------------------
// Phase 2: per-edge  e = relu(A[dst] - A[src] + B[src]); scatter-max to
// agg[src]. One wave per edge (uniform node index -> scalar loads), lane
// covers 8 channels. All values >= 0, so f32 max == u32 max on bit pattern:
// native global_atomic_max_u32, skipped when e == 0 (agg pre-zeroed).
// ---------------------------------------------------------------------------
__global__ __launch_bounds__(256)
void edge_scatter_max(const float* __restrict__ A,
                      const float* __restrict__ B,
                      const int* __restrict__ src,
                      const int* __restrict__ dst,
                      unsigned int* __restrict__ agg_bits) {
    const int wave = (blockIdx.x * blockDim.x + threadIdx.x) >> 5;
    const int lane = threadIdx.x & 31;
    if (wave >= N_EDGES) return;
    const int s = src[wave];
    const int d = dst[wave];
    const float* As = A + (size_t)s * C_OUT;
    const float* Ad = A + (size_t)d * C_OUT;
    const float* Bs = B + (size_t)s * C_OUT;
    unsigned int* out = agg_bits + (size_t)s * C_OUT;
#pragma unroll
    for (int j = 0; j < C_OUT / 32; ++j) {
        const int c = lane + 32 * j;
        const float e = fmaxf(Ad[c] - As[c] + Bs[c], 0.f);
        if (e > 0.f) atomicMax(out + c, __float_as_uint(e));
    }
}

// ---------------------------------------------------------------------------
// Phase 3: per-channel sum / sumsq partials (thread = channel -> coalesced),
// atomic f32 adds into sums[0..255]=sum, sums[256..511]=sumsq.
// ---------------------------------------------------------------------------
__global__ __launch_bounds__(256)
void bn_stats(const float* __restrict__ agg, float* __restrict__ sums) {
    const int c = threadIdx.x;
    const int rows_per_block = (N_NODES + gridDim.x - 1) / gridDim.x;
    const int r0 = blockIdx.x * rows_per_block;
    const int r1 = min(r0 + rows_per_block, N_NODES);
    float s = 0.f, ss = 0.f;
    for (int r = r0; r < r1; ++r) {
        const float v = agg[(size_t)r * C_OUT + c];
        s += v;
        ss += v * v;
    }
    atomicAdd(&sums[c], s);
    atomicAdd(&sums[C_OUT + c], ss);
}

// ---------------------------------------------------------------------------
// Phase 4: y = relu((agg - mean) * rsqrt(var + eps) * gamma + beta)
// mean = sum/N, var = sumsq/N - mean^2 (biased, matches jnp.var)
// ---------------------------------------------------------------------------
__global__ __launch_bounds__(256)
void bn_apply(const float* __restrict__ agg,
              const float* __restrict__ sums,
              const float* __restrict__ gamma,
              const float* __restrict__ beta,
              float* __restrict__ y) {
    const long long i = (long long)blockIdx.x * blockDim.x + threadIdx.x;
    if (i >= (long long)N_NODES * C_OUT) return;
    const int c = (int)(i & (C_OUT - 1));
    const float inv_n = 1.f / (float)N_NODES;
    const float mean = sums[c] * inv_n;
    const float var  = sums[C_OUT + c] * inv_n - mean * mean;
    const float v = (agg[i] - mean) * rsqrtf(var + BN_EPS) * gamma[c] + beta[c];
    y[i] = fmaxf(v, 0.f);
}

// ---------------------------------------------------------------------------
extern "C" void kernel_launch(void* const* d_in, const int* in_sizes, int n_in,
                              void* d_out, int out_size, void* d_ws, size_t ws_size,
                              hipStream_t stream) {
    const float* x       = (const float*)d_in[0];
    const int*   src     = (const int*)d_in[1];
    const int*   dst     = (const int*)d_in[2];
    const float* theta_w = (const float*)d_in[3];
    const float* phi_w   = (const float*)d_in[4];
    const float* gamma   = (const float*)d_in[5];
    const float* beta    = (const float*)d_in[6];
    float* y = (float*)d_out;

    const size_t nc = (size_t)N_NODES * C_OUT;   // 12.8M floats
    float* A    = (float*)d_ws;                  // [N, C_OUT]
    float* B    = A + nc;                        // [N, C_OUT]
    float* agg  = B + nc;                        // [N, C_OUT]
    float* sums = agg + nc;                      // [2 * C_OUT]

    // Phase 0: zero agg + sums (contiguous, multiple of 4 floats)
    {
        const long long n4 = (long long)(nc + 2 * C_OUT) / 4;
        const int blocks = (int)((n4 + 255) / 256);
        zero_kernel<<<blocks, 256, 0, stream>>>((float4*)agg, n4);
    }

    // Phase 1: node projections (50000 tiles, 4 waves/block -> 12500 blocks)
    node_proj_wmma<<<12500, 128, 0, stream>>>(x, theta_w, phi_w, A, B);

    // Phase 2: edge scatter-max (800000 waves, 8 waves/block -> 100000 blocks)
    edge_scatter_max<<<N_EDGES / 8, 256, 0, stream>>>(A, B, src, dst,
                                                      (unsigned int*)agg);

    // Phase 3: BN statistics
    bn_stats<<<128, C_OUT, 0, stream>>>(agg, sums);

    // Phase 4: normalize + relu -> d_out
    {
        const long long total = (long long)nc;
        const int blocks = (int)((total + 255) / 256);
        bn_apply<<<blocks, 256, 0, stream>>>(agg, sums, gamma, beta, y);
    }
}